// ExpEig_17789754540870
// MI455X (gfx1250) — compile-verified
//
#include <hip/hip_runtime.h>
#include <math.h>

typedef __attribute__((ext_vector_type(2))) float v2f;
typedef __attribute__((ext_vector_type(8))) float v8f;

#define MATN 128
// Row-pair interleaved LDS layout: element (i,j) at (i>>1)*SROW + 2*j + (i&1).
// SROW % 64 == 34 -> A-fragment column reads are bank-conflict-free and
// B-fragment b64 reads are nearly conflict-free.
#define SROW 290
#define MAT_FLOATS (64 * SROW)      // 64 row-pairs -> 18560 floats

__device__ __forceinline__ int ld_addr(int i, int j) {
    return (i >> 1) * SROW + 2 * j + (i & 1);
}

struct Frag {
    v2f a;       // A 16x4 fragment
    v2f b[8];    // B 4x16 fragments for the wave's 8 tiles
};

__device__ __forceinline__ void load_frag(Frag& f,
                                          const float* __restrict__ asrc,
                                          const float* __restrict__ bsrc,
                                          int abase, int bcol2, int hl, int k0)
{
    const int ka = k0 + hl * 2;
    const int rb = (ka >> 1) * SROW;        // row-pair base for B
    f.a.x = asrc[abase + 2 * ka];           // (m, ka)
    f.a.y = asrc[abase + 2 * ka + 2];       // (m, ka+1)
#pragma unroll
    for (int n = 0; n < 8; ++n)
        f.b[n] = *(const v2f*)&bsrc[rb + n * 32 + bcol2];   // one ds_load_b64
}

__device__ __forceinline__ void mma_frag(v8f acc[8], const Frag& f)
{
#pragma unroll
    for (int n = 0; n < 8; ++n)
        acc[n] = __builtin_amdgcn_wmma_f32_16x16x4_f32(
            false, f.a, false, f.b[n], (short)0, acc[n], false, false);
}

// C[16x16 tile] += A[16x4] * B[4x16] via V_WMMA_F32_16X16X4_F32.
// Each wave owns one 16-row band (8 tiles across); acc lives in VGPRs.
// Software-pipelined double-buffered fragment loads.
__device__ __forceinline__ void wave_matmul(const float* __restrict__ asrc,
                                            const float* __restrict__ bsrc,
                                            float* __restrict__ dst,
                                            float postscale, bool addI,
                                            int wave, int lane, const v8f& diagf)
{
    const int hl  = lane >> 4;      // lane half: 0 or 1
    const int l16 = lane & 15;
    const int m0  = wave * 16;      // this wave's row band

    v8f acc[8];
#pragma unroll
    for (int n = 0; n < 8; ++n) acc[n] = (v8f)0.0f;

    const int am    = m0 + l16;
    const int abase = (am >> 1) * SROW + (am & 1);
    const int bcol2 = 2 * l16;

    Frag f0, f1;
    load_frag(f0, asrc, bsrc, abase, bcol2, hl, 0);
    load_frag(f1, asrc, bsrc, abase, bcol2, hl, 4);
    // Steady state: consume one set while the next two loads are in flight.
    for (int k0 = 0; k0 < MATN - 8; k0 += 8) {
        mma_frag(acc, f0);
        load_frag(f0, asrc, bsrc, abase, bcol2, hl, k0 + 8);
        mma_frag(acc, f1);
        load_frag(f1, asrc, bsrc, abase, bcol2, hl, k0 + 12);
    }
    mma_frag(acc, f0);   // k0 = 120
    mma_frag(acc, f1);   // k0 = 124

    __syncthreads();   // all waves done READING asrc/bsrc before anyone overwrites dst

    // Writeback: rows (p, p+1) are adjacent dwords in the interleaved layout ->
    // paired b64 stores. Diagonal +1 only touches tile n == wave (uniform branch).
#pragma unroll
    for (int n = 0; n < 8; ++n) {
        const int base = ((m0 + hl * 8) >> 1) * SROW + 2 * (n * 16 + l16);
        if (addI && n == wave) {
#pragma unroll
            for (int p = 0; p < 8; p += 2) {
                v2f w;
                w.x = fmaf(acc[n][p],     postscale, diagf[p]);
                w.y = fmaf(acc[n][p + 1], postscale, diagf[p + 1]);
                *(v2f*)&dst[base + (p >> 1) * SROW] = w;
            }
        } else {
#pragma unroll
            for (int p = 0; p < 8; p += 2) {
                v2f w;
                w.x = acc[n][p]     * postscale;
                w.y = acc[n][p + 1] * postscale;
                *(v2f*)&dst[base + (p >> 1) * SROW] = w;
            }
        }
    }
    __syncthreads();   // dst fully written before next stage reads it
}

__global__ void __launch_bounds__(256)
expm_scale_square_kernel(const float* __restrict__ x, float* __restrict__ out)
{
    extern __shared__ float smem[];
    float* Am  = smem;                        // scaled symmetrized input (A operand)
    float* P   = smem + MAT_FLOATS;           // working matrix
    float* red = smem + 2 * MAT_FLOATS;       // 128-float reduction scratch

    const int b    = blockIdx.x;
    const int tid  = threadIdx.x;
    const int wave = tid >> 5;
    const int lane = tid & 31;
    const int hl   = lane >> 4;
    const int l16  = lane & 15;
    const float* xb = x + (size_t)b * MATN * MATN;

    // Per-lane identity fragment for the wave's diagonal tile:
    // element (row = hl*8 + p, col = l16) of a 16x16 tile.
    v8f diagf;
#pragma unroll
    for (int p = 0; p < 8; ++p) diagf[p] = (l16 == hl * 8 + p) ? 1.0f : 0.0f;

    // Stage raw matrix into LDS with fully coalesced global reads (into P slot).
    for (int idx = tid; idx < MATN * MATN; idx += 256) {
        int i = idx >> 7, j = idx & 127;
        P[ld_addr(i, j)] = xb[idx];
    }
    __syncthreads();

    // Symmetrize from lower triangle (matches eigh UPLO='L') out of LDS.
    for (int idx = tid; idx < MATN * MATN; idx += 256) {
        int i = idx >> 7, j = idx & 127;
        Am[ld_addr(i, j)] = (i >= j) ? P[ld_addr(i, j)] : P[ld_addr(j, i)];
    }
    __syncthreads();

    // Infinity norm: max row sum of |A|.
    if (tid < MATN) {
        float s = 0.0f;
        int base = (tid >> 1) * SROW + (tid & 1);
        for (int j = 0; j < MATN; ++j) s += fabsf(Am[base + 2 * j]);
        red[tid] = s;
    }
    __syncthreads();
    if (tid == 0) {
        float m = 0.0f;
        for (int i = 0; i < MATN; ++i) m = fmaxf(m, red[i]);
        red[0] = m;
    }
    __syncthreads();

    // Scaling: bring ||B|| <= 1 so degree-12 Taylor error ~1e-10.
    float nrm = red[0];
    int s = 0;
    if (nrm > 1.0f) s = (int)ceilf(log2f(nrm));
    if (s > 30) s = 30;
    s = __builtin_amdgcn_readfirstlane(s);    // keep squaring loop scalar
    const float scale = exp2f(-(float)s);

    // Scale A and (raw data now consumed) re-init P = I.
    for (int idx = tid; idx < MATN * MATN; idx += 256) {
        int i = idx >> 7, j = idx & 127;
        int a = ld_addr(i, j);
        Am[a] *= scale;
        P [a] = (i == j) ? 1.0f : 0.0f;
    }
    __syncthreads();

    // Horner Taylor: P = I + (Am * P)/k, k = 12..1  ->  P = sum_{j<=12} Am^j/j!
    // Fully unrolled so 1/k folds to a literal (no runtime divide sequence).
#pragma unroll
    for (int k = 12; k >= 1; --k)
        wave_matmul(Am, P, P, 1.0f / (float)k, true, wave, lane, diagf);

    // Undo scaling: P <- P^(2^s). postscale==1, addI==false -> pure stores.
    for (int i = 0; i < s; ++i)
        wave_matmul(P, P, P, 1.0f, false, wave, lane, diagf);

    // Coalesced store.
    float* ob = out + (size_t)b * MATN * MATN;
    for (int idx = tid; idx < MATN * MATN; idx += 256) {
        int i = idx >> 7, j = idx & 127;
        ob[idx] = P[ld_addr(i, j)];
    }
}

extern "C" void kernel_launch(void* const* d_in, const int* in_sizes, int n_in,
                              void* d_out, int out_size, void* d_ws, size_t ws_size,
                              hipStream_t stream) {
    const float* x = (const float*)d_in[0];
    float* out = (float*)d_out;
    const int batches = in_sizes[0] / (MATN * MATN);       // 4096
    const size_t shmem = (size_t)(2 * MAT_FLOATS + 128) * sizeof(float); // ~145.5 KB
    hipLaunchKernelGGL(expm_scale_square_kernel, dim3(batches), dim3(256), shmem, stream,
                       x, out);
}